// GCNEncoder_7791070674960
// MI455X (gfx1250) — compile-verified
//
#include <hip/hip_runtime.h>

#define N_NODES 100000
#define N_EDGES 1600000
#define IN_CH 128
#define HID 64
#define EMB 32

typedef __attribute__((ext_vector_type(2))) float v2f;
typedef __attribute__((ext_vector_type(8))) float v8f;

// ---------------------------------------------------------------------------
// Degree / normalization
// ---------------------------------------------------------------------------
__global__ void k_deg_init(int* __restrict__ deg) {
    int i = blockIdx.x * blockDim.x + threadIdx.x;
    if (i < N_NODES) deg[i] = 1;  // self-loop contributes 1
}

__global__ void k_deg_edges(const int* __restrict__ dst, int* __restrict__ deg) {
    int e = blockIdx.x * blockDim.x + threadIdx.x;
    if (e < N_EDGES) atomicAdd(&deg[dst[e]], 1);
}

__global__ void k_dinv(const int* __restrict__ deg, float* __restrict__ dinv) {
    int i = blockIdx.x * blockDim.x + threadIdx.x;
    if (i < N_NODES) dinv[i] = rsqrtf((float)deg[i]);  // deg >= 1 always
}

// ---------------------------------------------------------------------------
// GEMM 1: h1lin[N,64] = x[N,128] @ W1[128,64]  (f32 WMMA 16x16x4, wave32)
// One wave per 16x16 output tile; W1 staged in LDS.
// ---------------------------------------------------------------------------
__global__ __launch_bounds__(256) void k_gemm1(const float* __restrict__ x,
                                               const float* __restrict__ W1,
                                               float* __restrict__ h1lin) {
    __shared__ float w1s[IN_CH * HID];  // 32 KB
    for (int i = threadIdx.x; i < IN_CH * HID / 4; i += 256)
        ((float4*)w1s)[i] = ((const float4*)W1)[i];
    __syncthreads();

    const int wave = threadIdx.x >> 5;
    const int lane = threadIdx.x & 31;
    const int tile = blockIdx.x * 8 + wave;          // 25000 tiles exactly
    const int tm = tile >> 2;                        // N/16 row tiles
    const int tn = tile & 3;                         // HID/16 = 4 col tiles
    const int row0 = tm * 16, col0 = tn * 16;
    const int half = lane >> 4, l16 = lane & 15;

    // A layout (16x4 f32): lanes 0-15 -> K={k,k+1}; lanes 16-31 -> K={k+2,k+3}
    const float* arow = x + (size_t)(row0 + l16) * IN_CH + 2 * half;
    // B layout (4x16 f32): row K striped across lanes within a VGPR
    const float* bcol = w1s + (size_t)(2 * half) * HID + col0 + l16;

    v8f c = {};
#pragma unroll
    for (int k = 0; k < IN_CH; k += 4) {
        v2f a = *(const v2f*)(arow + k);             // global_load_b64
        v2f b;
        b.x = bcol[k * HID];                          // ds_load
        b.y = bcol[k * HID + HID];
        c = __builtin_amdgcn_wmma_f32_16x16x4_f32(false, a, false, b,
                                                  (short)0, c, false, false);
    }

    // C/D layout: VGPR r -> (M = r + 8*half, N = l16)
    float* out = h1lin + (size_t)(row0 + 8 * half) * HID + col0 + l16;
#pragma unroll
    for (int r = 0; r < 8; ++r) out[(size_t)r * HID] = c[r];
}

// ---------------------------------------------------------------------------
// GEMM 2+3 fused: h2lin = relu(h1)@W2, h3lin = relu(h1)@W3  (shared A frags)
// ---------------------------------------------------------------------------
__global__ __launch_bounds__(256) void k_gemm23(const float* __restrict__ h1,
                                                const float* __restrict__ W2,
                                                const float* __restrict__ W3,
                                                float* __restrict__ h2lin,
                                                float* __restrict__ h3lin) {
    __shared__ float w2s[HID * EMB];  // 8 KB
    __shared__ float w3s[HID * EMB];  // 8 KB
    for (int i = threadIdx.x; i < HID * EMB / 4; i += 256) {
        ((float4*)w2s)[i] = ((const float4*)W2)[i];
        ((float4*)w3s)[i] = ((const float4*)W3)[i];
    }
    __syncthreads();

    const int wave = threadIdx.x >> 5;
    const int lane = threadIdx.x & 31;
    const int tile = blockIdx.x * 8 + wave;
    if (tile >= (N_NODES / 16) * (EMB / 16)) return;  // 12500 tiles
    const int tm = tile >> 1;
    const int tn = tile & 1;                          // EMB/16 = 2 col tiles
    const int row0 = tm * 16, col0 = tn * 16;
    const int half = lane >> 4, l16 = lane & 15;

    const float* arow = h1 + (size_t)(row0 + l16) * HID + 2 * half;
    const float* b2c = w2s + (size_t)(2 * half) * EMB + col0 + l16;
    const float* b3c = w3s + (size_t)(2 * half) * EMB + col0 + l16;

    v8f c2 = {}, c3 = {};
#pragma unroll
    for (int k = 0; k < HID; k += 4) {
        v2f a = *(const v2f*)(arow + k);
        a.x = fmaxf(a.x, 0.0f);                       // fused ReLU on h1
        a.y = fmaxf(a.y, 0.0f);
        v2f b2; b2.x = b2c[k * EMB]; b2.y = b2c[k * EMB + EMB];
        v2f b3; b3.x = b3c[k * EMB]; b3.y = b3c[k * EMB + EMB];
        c2 = __builtin_amdgcn_wmma_f32_16x16x4_f32(false, a, false, b2,
                                                   (short)0, c2, false, false);
        c3 = __builtin_amdgcn_wmma_f32_16x16x4_f32(false, a, false, b3,
                                                   (short)0, c3, false, false);
    }

    float* o2 = h2lin + (size_t)(row0 + 8 * half) * EMB + col0 + l16;
    float* o3 = h3lin + (size_t)(row0 + 8 * half) * EMB + col0 + l16;
#pragma unroll
    for (int r = 0; r < 8; ++r) {
        o2[(size_t)r * EMB] = c2[r];
        o3[(size_t)r * EMB] = c3[r];
    }
}

// ---------------------------------------------------------------------------
// Aggregation layer 1: init with self-loop + bias, then edge scatter-add
// ---------------------------------------------------------------------------
__global__ void k_selfbias1(const float* __restrict__ h1lin,
                            const float* __restrict__ dinv,
                            const float* __restrict__ b1,
                            float* __restrict__ h1) {
    int idx = blockIdx.x * blockDim.x + threadIdx.x;
    if (idx >= N_NODES * HID) return;
    int i = idx >> 6, c = idx & 63;
    float d = dinv[i];
    h1[idx] = d * d * h1lin[idx] + b1[c];
}

__global__ void k_agg_edges1(const int* __restrict__ src,
                             const int* __restrict__ dst,
                             const float* __restrict__ dinv,
                             const float* __restrict__ h1lin,
                             float* __restrict__ h1) {
    int tid = blockIdx.x * blockDim.x + threadIdx.x;  // 16 threads per edge
    int e = tid >> 4;
    if (e >= N_EDGES) return;
    int c4 = (tid & 15) << 2;                         // 4 channels per thread
    int s = src[e], d = dst[e];
    float w = dinv[s] * dinv[d];
    float4 v = *(const float4*)(h1lin + (size_t)s * HID + c4);
    float* o = h1 + (size_t)d * HID + c4;
    unsafeAtomicAdd(o + 0, w * v.x);                  // global_atomic_add_f32
    unsafeAtomicAdd(o + 1, w * v.y);
    unsafeAtomicAdd(o + 2, w * v.z);
    unsafeAtomicAdd(o + 3, w * v.w);
}

// ---------------------------------------------------------------------------
// Aggregation layers 2+3 (writes directly to d_out: out1 | out2)
// ---------------------------------------------------------------------------
__global__ void k_selfbias23(const float* __restrict__ h2lin,
                             const float* __restrict__ h3lin,
                             const float* __restrict__ dinv,
                             const float* __restrict__ b2,
                             const float* __restrict__ b3,
                             float* __restrict__ out1,
                             float* __restrict__ out2) {
    int idx = blockIdx.x * blockDim.x + threadIdx.x;
    if (idx >= N_NODES * 2 * EMB) return;
    int i = idx >> 6, c = idx & 63;
    float d = dinv[i];
    float dd = d * d;
    if (c < EMB)
        out1[(size_t)i * EMB + c] = dd * h2lin[(size_t)i * EMB + c] + b2[c];
    else
        out2[(size_t)i * EMB + (c - EMB)] =
            dd * h3lin[(size_t)i * EMB + (c - EMB)] + b3[c - EMB];
}

__global__ void k_agg_edges23(const int* __restrict__ src,
                              const int* __restrict__ dst,
                              const float* __restrict__ dinv,
                              const float* __restrict__ h2lin,
                              const float* __restrict__ h3lin,
                              float* __restrict__ out1,
                              float* __restrict__ out2) {
    int tid = blockIdx.x * blockDim.x + threadIdx.x;  // 16 threads per edge
    int e = tid >> 4;
    if (e >= N_EDGES) return;
    int q = tid & 15;
    int s = src[e], d = dst[e];
    float w = dinv[s] * dinv[d];
    const float* hl;
    float* o;
    int c4;
    if (q < 8) { c4 = q << 2; hl = h2lin; o = out1; }
    else       { c4 = (q - 8) << 2; hl = h3lin; o = out2; }
    float4 v = *(const float4*)(hl + (size_t)s * EMB + c4);
    float* p = o + (size_t)d * EMB + c4;
    unsafeAtomicAdd(p + 0, w * v.x);
    unsafeAtomicAdd(p + 1, w * v.y);
    unsafeAtomicAdd(p + 2, w * v.z);
    unsafeAtomicAdd(p + 3, w * v.w);
}

// ---------------------------------------------------------------------------
extern "C" void kernel_launch(void* const* d_in, const int* in_sizes, int n_in,
                              void* d_out, int out_size, void* d_ws, size_t ws_size,
                              hipStream_t stream) {
    const float* x     = (const float*)d_in[0];
    const int*   edges = (const int*)  d_in[1];   // [2, E] row-major
    const float* W1    = (const float*)d_in[2];
    const float* b1    = (const float*)d_in[3];
    const float* W2    = (const float*)d_in[4];
    const float* b2    = (const float*)d_in[5];
    const float* W3    = (const float*)d_in[6];
    const float* b3    = (const float*)d_in[7];
    const int* src = edges;
    const int* dst = edges + N_EDGES;

    char* ws = (char*)d_ws;
    size_t off = 0;
    auto carve = [&](size_t bytes) -> void* {
        void* p = ws + off;
        off = (off + bytes + 255) & ~(size_t)255;
        return p;
    };
    int*   deg   = (int*)  carve((size_t)N_NODES * 4);
    float* dinv  = (float*)carve((size_t)N_NODES * 4);
    float* h1lin = (float*)carve((size_t)N_NODES * HID * 4);
    float* h1    = (float*)carve((size_t)N_NODES * HID * 4);
    float* h2lin = (float*)carve((size_t)N_NODES * EMB * 4);
    float* h3lin = (float*)carve((size_t)N_NODES * EMB * 4);

    float* out1 = (float*)d_out;
    float* out2 = out1 + (size_t)N_NODES * EMB;

    const int TB = 256;
    // 1) degrees + dinv
    k_deg_init <<<(N_NODES + TB - 1) / TB, TB, 0, stream>>>(deg);
    k_deg_edges<<<(N_EDGES + TB - 1) / TB, TB, 0, stream>>>(dst, deg);
    k_dinv     <<<(N_NODES + TB - 1) / TB, TB, 0, stream>>>(deg, dinv);

    // 2) h1lin = x @ W1 (WMMA); 25000 tiles, 8 waves/block
    k_gemm1<<<(N_NODES / 16) * (HID / 16) / 8, TB, 0, stream>>>(x, W1, h1lin);

    // 3) h1 = Agg(h1lin) + b1  (ReLU deferred to GEMM2/3 A-load)
    k_selfbias1<<<(N_NODES * HID + TB - 1) / TB, TB, 0, stream>>>(h1lin, dinv, b1, h1);
    k_agg_edges1<<<(N_EDGES * 16 + TB - 1) / TB, TB, 0, stream>>>(src, dst, dinv, h1lin, h1);

    // 4) h2lin/h3lin = relu(h1) @ {W2, W3}; 12500 tiles
    int tiles23 = (N_NODES / 16) * (EMB / 16);
    k_gemm23<<<(tiles23 + 7) / 8, TB, 0, stream>>>(h1, W2, W3, h2lin, h3lin);

    // 5) out1/out2 = Agg(h2lin/h3lin) + b2/b3
    k_selfbias23<<<(N_NODES * 2 * EMB + TB - 1) / TB, TB, 0, stream>>>(
        h2lin, h3lin, dinv, b2, b3, out1, out2);
    k_agg_edges23<<<(N_EDGES * 16 + TB - 1) / TB, TB, 0, stream>>>(
        src, dst, dinv, h2lin, h3lin, out1, out2);
}